// AlphaRankingLoss_76424648065071
// MI455X (gfx1250) — compile-verified
//
#include <hip/hip_runtime.h>

#define ALPHA_C  0.7f
#define MARGIN_C 0.5f
#define BATCH_N  8192
#define THREADS  256
#define SPLIT    8
#define SLICE    (BATCH_N / SPLIT)     // 1024 columns per block
#define NBLOCKS  ((BATCH_N / THREADS) * SPLIT)   // 256 partial slots

#if defined(__AMDGCN__) && __has_builtin(__builtin_amdgcn_global_load_async_to_lds_b128)
#define USE_ASYNC_LDS 1
#endif

typedef int v4i __attribute__((vector_size(16)));
typedef __attribute__((address_space(1))) v4i* v4i_gptr;   // global, 64-bit
typedef __attribute__((address_space(3))) v4i* v4i_lptr;   // LDS, 32-bit

// ---- async global -> LDS staging (gfx1250 path), with portable fallback ----
__device__ __forceinline__ void stage_b128(const float* g, float* l) {
#ifdef USE_ASYNC_LDS
  v4i_gptr gp = (v4i_gptr)(unsigned long long)g;
  v4i_lptr lp = (v4i_lptr)(unsigned)(unsigned long long)l;  // low 32 bits = LDS offset
  __builtin_amdgcn_global_load_async_to_lds_b128(gp, lp, /*offset=*/0, /*cpol=*/0);
#else
  *(float4*)l = *(const float4*)g;
#endif
}

__device__ __forceinline__ void stage_wait() {
#ifdef USE_ASYNC_LDS
#if __has_builtin(__builtin_amdgcn_s_wait_asynccnt)
  __builtin_amdgcn_s_wait_asynccnt(0);
#else
  asm volatile("s_wait_asynccnt 0" ::: "memory");
#endif
#endif
}

// ---- workspace layout (all slots rewritten every launch; no init needed) ----
// ws[0            .. NBLOCKS-1 ]  : per-block ranking sums   (f32)
// ws[NBLOCKS      .. 2*NBLOCKS-1] : per-block mse sums       (f32)
// ws[2*NBLOCKS    .. 3*NBLOCKS-1] : per-block pair counts    (u32 bits)

__global__ void rank_loss_kernel(const float* __restrict__ p,
                                 const float* __restrict__ t,
                                 float* __restrict__ ws) {
  __shared__ float    t_sl[SLICE];
  __shared__ float    p_sl[SLICE];
  __shared__ float    red_s[THREADS];
  __shared__ float    red_m[THREADS];
  __shared__ unsigned red_c[THREADS];

  const int tid  = threadIdx.x;
  const int rb   = blockIdx.x;          // row block    [0,32)
  const int cb   = blockIdx.y;          // column split [0,8)
  const int col0 = cb * SLICE;
  const int bid  = cb * gridDim.x + rb; // 0..255 partial slot

  // one async b128 per thread per array fills the whole 1024-wide slice
  stage_b128(t + col0 + tid * 4, t_sl + tid * 4);
  stage_b128(p + col0 + tid * 4, p_sl + tid * 4);

  const int   i  = rb * THREADS + tid;  // this thread's row
  const float ti = t[i];
  const float pi = p[i];

  stage_wait();
  __syncthreads();

  float    sum = 0.0f;
  unsigned cnt = 0u;

  const float4* t4 = (const float4*)t_sl;
  const float4* p4 = (const float4*)p_sl;

#define PAIR(tj, pj)                                      \
  {                                                       \
    float dt  = ti - (tj);                                \
    float dp  = pi - (pj);                                \
    float sdp = (dt > 0.0f) ? dp : -dp;                   \
    float lv  = MARGIN_C - sdp;                           \
    lv        = (lv > 0.0f) ? lv : 0.0f;                  \
    bool  ok  = (dt != 0.0f);                             \
    sum += ok ? lv : 0.0f;                                \
    cnt += ok ? 1u : 0u;                                  \
  }

#pragma unroll 4
  for (int j = 0; j < SLICE / 4; ++j) {
    float4 tv = t4[j];
    float4 pv = p4[j];
    PAIR(tv.x, pv.x)
    PAIR(tv.y, pv.y)
    PAIR(tv.z, pv.z)
    PAIR(tv.w, pv.w)
  }
#undef PAIR

  // MSE term: each row exactly once (only in column-split 0)
  float d    = pi - ti;
  float msep = (cb == 0) ? d * d : 0.0f;

  red_s[tid] = sum;
  red_m[tid] = msep;
  red_c[tid] = cnt;
  __syncthreads();

  for (int off = THREADS / 2; off > 0; off >>= 1) {
    if (tid < off) {
      red_s[tid] += red_s[tid + off];
      red_m[tid] += red_m[tid + off];
      red_c[tid] += red_c[tid + off];
    }
    __syncthreads();
  }

  if (tid == 0) {
    ws[bid]                = red_s[0];
    ws[NBLOCKS + bid]      = red_m[0];
    ((unsigned*)ws)[2 * NBLOCKS + bid] = red_c[0];
  }
}

// single 256-thread block: fixed-order tree reduction -> bit-exact across replays
__global__ void finalize_kernel(const float* __restrict__ ws,
                                float* __restrict__ out) {
  __shared__ float    red_s[NBLOCKS];
  __shared__ float    red_m[NBLOCKS];
  __shared__ unsigned red_c[NBLOCKS];

  const int tid = threadIdx.x;
  red_s[tid] = ws[tid];
  red_m[tid] = ws[NBLOCKS + tid];
  red_c[tid] = ((const unsigned*)ws)[2 * NBLOCKS + tid];
  __syncthreads();

  for (int off = NBLOCKS / 2; off > 0; off >>= 1) {
    if (tid < off) {
      red_s[tid] += red_s[tid + off];
      red_m[tid] += red_m[tid + off];
      red_c[tid] += red_c[tid + off];
    }
    __syncthreads();
  }

  if (tid == 0) {
    float    rsum = red_s[0];
    float    msum = red_m[0];
    unsigned cnt  = red_c[0];

    float mse  = msum * (1.0f / (float)BATCH_N);
    float rank = (cnt > 0u) ? (rsum / (float)cnt) : 0.0f;
    out[0]     = ALPHA_C * mse + (1.0f - ALPHA_C) * rank;
  }
}

extern "C" void kernel_launch(void* const* d_in, const int* in_sizes, int n_in,
                              void* d_out, int out_size, void* d_ws, size_t ws_size,
                              hipStream_t stream) {
  const float* p   = (const float*)d_in[0];
  const float* t   = (const float*)d_in[1];
  float*       out = (float*)d_out;
  float*       ws  = (float*)d_ws;

  dim3 grid(BATCH_N / THREADS, SPLIT);   // 32 x 8 = 256 workgroups
  rank_loss_kernel<<<grid, THREADS, 0, stream>>>(p, t, ws);

  finalize_kernel<<<1, NBLOCKS, 0, stream>>>(ws, out);
}